// Generator_9758165696833
// MI455X (gfx1250) — compile-verified
//
#include <hip/hip_runtime.h>

#define Bz 32
#define Dz 64
#define Nz 254
#define Lz 16
#define Hz 128
#define Pz 96

typedef __attribute__((ext_vector_type(2))) float v2f;
typedef __attribute__((ext_vector_type(8))) float v8f;

// ---------------------------------------------------------------------------
// Kernel 0: Spe[h] = sum_n pos_embed[n][h]   (128 floats, one tiny block)
// ---------------------------------------------------------------------------
__global__ void spe_kernel(const float* __restrict__ pe, float* __restrict__ Spe) {
    int h = threadIdx.x;                // 128 threads
    float s = 0.f;
    for (int n = 0; n < Nz; ++n) s += pe[n * Hz + h];
    Spe[h] = s;
}

// ---------------------------------------------------------------------------
// Kernel 1: per-(b,d) attention-pool using the algebraic fold of the big GEMM.
// One 256-thread block per (b,d). Writes pooled (2048 x 16) to workspace.
// ---------------------------------------------------------------------------
__global__ void __launch_bounds__(256) pooled_kernel(
    const float* __restrict__ x,      // (B,D,N,L)
    const float* __restrict__ W_enc,  // (L,H)
    const float* __restrict__ b_enc,  // (H)
    const float* __restrict__ pe,     // (N,H)
    const float* __restrict__ Spe,    // (H)
    float* __restrict__ pooled)       // (B*D, L)
{
    __shared__ float sx[Nz * Lz];     // 4064 floats: this channel's patches
    __shared__ float sxs[Lz];         // column sums of x
    __shared__ float scsum[Hz];       // csum vector
    __shared__ float sv[Lz];          // W_enc @ csum
    __shared__ float sbeta;           // b_enc . csum
    __shared__ float sscore[Nz];      // raw scores
    __shared__ float sred[256];       // reduction scratch

    const int bd = blockIdx.x;
    const int t  = threadIdx.x;
    const float* xb = x + (size_t)bd * (Nz * Lz);

    for (int i = t; i < Nz * Lz; i += 256) sx[i] = xb[i];
    __syncthreads();

    // xs[l] = sum_n x[n][l]   (16 lanes x 16 groups)
    const int l = t & 15, g = t >> 4;
    {
        float part = 0.f;
        for (int n = g; n < Nz; n += 16) part += sx[n * Lz + l];
        sred[t] = part;
    }
    __syncthreads();
    if (t < Lz) {
        float s = 0.f;
        for (int gg = 0; gg < 16; ++gg) s += sred[gg * 16 + t];
        sxs[t] = s;
    }
    __syncthreads();

    // csum[h] = xs @ W_enc + N*b_enc[h] + Spe[h]
    if (t < Hz) {
        float s = (float)Nz * b_enc[t] + Spe[t];
        for (int ll = 0; ll < Lz; ++ll) s += sxs[ll] * W_enc[ll * Hz + t];
        scsum[t] = s;
    }
    __syncthreads();

    // v[l] = W_enc[l] . csum ;  beta = b_enc . csum
    if (t < Lz) {
        float s = 0.f;
        for (int h = 0; h < Hz; ++h) s += W_enc[t * Hz + h] * scsum[h];
        sv[t] = s;
    }
    if (t == 16) {
        float s = 0.f;
        for (int h = 0; h < Hz; ++h) s += b_enc[h] * scsum[h];
        sbeta = s;
    }
    __syncthreads();

    // score_i = x_i.v + beta + pe_i.csum
    if (t < Nz) {
        float gamma = 0.f;
        const float* per = pe + t * Hz;
        for (int h = 0; h < Hz; ++h) gamma += per[h] * scsum[h];
        float xv = 0.f;
        for (int ll = 0; ll < Lz; ++ll) xv += sx[t * Lz + ll] * sv[ll];
        sscore[t] = xv + sbeta + gamma;
    }
    __syncthreads();

    // denom = sum_i |score_i|
    sred[t] = (t < Nz) ? fabsf(sscore[t]) : 0.f;
    __syncthreads();
    for (int off = 128; off >= 1; off >>= 1) {
        if (t < off) sred[t] += sred[t + off];
        __syncthreads();
    }
    const float denom = sred[0];
    __syncthreads();

    // pooled[l] = (sum_i x[i][l] * score_i) / denom
    {
        float part = 0.f;
        for (int n = g; n < Nz; n += 16) part += sx[n * Lz + l] * sscore[n];
        sred[t] = part;
    }
    __syncthreads();
    if (t < Lz) {
        float s = 0.f;
        for (int gg = 0; gg < 16; ++gg) s += sred[gg * 16 + t];
        pooled[bd * Lz + t] = s / denom;
    }
}

// ---------------------------------------------------------------------------
// Kernel 2: FFN via fp32 WMMA (V_WMMA_F32_16X16X4_F32).
//   h   = leaky_relu(pooled(2048x16) @ W1(16x128) + b1)
//   out = h @ W2(128x96) + b2, written transposed as (B, P, D)
// 8 waves / block, each wave owns a 16-row tile of `pooled`.
// A-frag layout (16x4 f32): lanes 0-15 -> K=0,1 ; lanes 16-31 -> K=2,3.
// C/D layout: VGPR r = row (r | (lane<16?0:8)), col = lane&15.
// ---------------------------------------------------------------------------
__global__ void __launch_bounds__(256) ffn_kernel(
    const float* __restrict__ pooled, // (2048, 16)
    const float* __restrict__ W1,     // (16, 128)
    const float* __restrict__ b1,     // (128)
    const float* __restrict__ W2,     // (128, 96)
    const float* __restrict__ b2,     // (96)
    float* __restrict__ out)          // (B, P, D)
{
    __shared__ float hl[8 * 16 * Hz]; // 8 waves x (16 x 128) hidden tiles = 64KB

    const int lane    = threadIdx.x & 31;
    const int wave    = threadIdx.x >> 5;
    const int rowbase = (blockIdx.x * 8 + wave) * 16;
    float* hw = hl + wave * (16 * Hz);

    const int m     = lane & 15;              // row (A) / col (B,D) index
    const int ksel  = (lane >> 4) << 1;       // 0 for lanes 0-15, 2 for 16-31
    const int mbase = (lane < 16) ? 0 : 8;    // D-frag row base

    // ---- GEMM1: 16x16 @ 16x128 -> h, with bias + leaky_relu(0.2) ----
    const float* arow = pooled + (size_t)(rowbase + m) * Lz;
    for (int ht = 0; ht < 8; ++ht) {
        v8f acc = {};
        const int col = ht * 16 + m;
        for (int kc = 0; kc < 4; ++kc) {
            const int k = kc * 4 + ksel;
            v2f a  = { arow[k], arow[k + 1] };
            v2f bf = { W1[k * Hz + col], W1[(k + 1) * Hz + col] };
            acc = __builtin_amdgcn_wmma_f32_16x16x4_f32(
                false, a, false, bf, (short)0, acc, false, false);
        }
        const float bias = b1[col];
        for (int r = 0; r < 8; ++r) {
            float v = acc[r] + bias;
            v = (v > 0.f) ? v : 0.2f * v;
            hw[(mbase + r) * Hz + col] = v;
        }
    }
    __syncthreads();

    // ---- GEMM2: 16x128 @ 128x96 -> out tile, bias, transposed store ----
    for (int tt = 0; tt < 6; ++tt) {
        v8f acc = {};
        const int col = tt * 16 + m;
        for (int kc = 0; kc < 32; ++kc) {
            const int k = kc * 4 + ksel;
            v2f a  = { hw[m * Hz + k], hw[m * Hz + k + 1] };
            v2f bf = { W2[k * Pz + col], W2[(k + 1) * Pz + col] };
            acc = __builtin_amdgcn_wmma_f32_16x16x4_f32(
                false, a, false, bf, (short)0, acc, false, false);
        }
        const float bias = b2[col];
        for (int r = 0; r < 8; ++r) {
            const int grow = rowbase + mbase + r;   // global (b*D + d) row
            const int bb = grow >> 6;               // / D
            const int dd = grow & 63;               // % D
            out[((size_t)bb * Pz + col) * Dz + dd] = acc[r] + bias;
        }
    }
}

// ---------------------------------------------------------------------------
extern "C" void kernel_launch(void* const* d_in, const int* in_sizes, int n_in,
                              void* d_out, int out_size, void* d_ws, size_t ws_size,
                              hipStream_t stream) {
    const float* x     = (const float*)d_in[0];
    const float* W_enc = (const float*)d_in[1];
    const float* b_enc = (const float*)d_in[2];
    const float* W1    = (const float*)d_in[3];
    const float* b1    = (const float*)d_in[4];
    const float* W2    = (const float*)d_in[5];
    const float* b2    = (const float*)d_in[6];
    const float* pe    = (const float*)d_in[7];

    float* out = (float*)d_out;

    float* Spe    = (float*)d_ws;       // 128 floats
    float* pooled = Spe + 128;          // 2048*16 floats

    spe_kernel<<<1, 128, 0, stream>>>(pe, Spe);
    pooled_kernel<<<Bz * Dz, 256, 0, stream>>>(x, W_enc, b_enc, pe, Spe, pooled);
    ffn_kernel<<<(Bz * Dz) / (16 * 8), 256, 0, stream>>>(pooled, W1, b1, W2, b2, out);

    // second output: p = x (exact copy), appended after out (B*P*D floats)
    hipMemcpyAsync(out + (size_t)Bz * Pz * Dz, x,
                   (size_t)Bz * Dz * Nz * Lz * sizeof(float),
                   hipMemcpyDeviceToDevice, stream);
}